// SwinTransformer_20590073217243
// MI455X (gfx1250) — compile-verified
//
#include <hip/hip_runtime.h>
#include <math.h>

// ---------------- problem constants ----------------
#define Bz     32
#define Hh     56
#define Wy     56
#define Cc     256
#define NHEADS 8
#define WSZ    7
#define SHIFT  3
#define MLPD   1024
#define NTOK   49            // tokens per window
#define HD     32            // head dim
#define NWIN   64            // windows per image (8x8)
#define TOKENS (Bz*Hh*Wy)    // 100352
#define SCALE  0.17677669529663687f   // 32^-0.5

typedef __attribute__((ext_vector_type(2))) float v2f;
typedef __attribute__((ext_vector_type(8))) float v8f;

__device__ __forceinline__ v8f wmma_f32(v2f a, v2f b, v8f c) {
  return __builtin_amdgcn_wmma_f32_16x16x4_f32(false, a, false, b, (short)0, c,
                                               false, false);
}

// region id for the shifted-window attention mask (computed on original grid)
__device__ __forceinline__ int region_id(int h, int w) {
  int rh = (h < Hh - WSZ) ? 0 : ((h < Hh - SHIFT) ? 1 : 2);
  int rw = (w < Wy - WSZ) ? 0 : ((w < Wy - SHIFT) ? 1 : 2);
  return rh * 3 + rw;
}

// ---------------------------------------------------------------------------
// LayerNorm over C=256, one wave (32 lanes) per token, 8 channels per lane.
// shifted=1: output in shifted+window-partitioned token order (reads rolled x).
// ---------------------------------------------------------------------------
__global__ __launch_bounds__(256) void swin_ln_kernel(
    const float* __restrict__ x, const float* __restrict__ g,
    const float* __restrict__ be, float* __restrict__ y, int shifted) {
  int wave = threadIdx.x >> 5;
  int lane = threadIdx.x & 31;
  int tok = blockIdx.x * 8 + wave;          // output token index
  long src;
  if (shifted) {
    int win = tok / NTOK, t = tok % NTOK;
    int b = win / NWIN, w2 = win % NWIN;
    int wh = w2 >> 3, ww = w2 & 7;
    int i = t / WSZ, j = t % WSZ;
    int ho = (wh * WSZ + i + SHIFT) % Hh;
    int wo = (ww * WSZ + j + SHIFT) % Wy;
    src = (long)b * (Hh * Wy) + ho * Wy + wo;
  } else {
    src = tok;
  }
  int c = lane * 8;
  const float* xp = x + src * Cc + c;
  float4 r0 = *(const float4*)(xp);
  float4 r1 = *(const float4*)(xp + 4);
  float s  = r0.x + r0.y + r0.z + r0.w + r1.x + r1.y + r1.z + r1.w;
  float s2 = r0.x*r0.x + r0.y*r0.y + r0.z*r0.z + r0.w*r0.w +
             r1.x*r1.x + r1.y*r1.y + r1.z*r1.z + r1.w*r1.w;
  #pragma unroll
  for (int o = 16; o >= 1; o >>= 1) {
    s  += __shfl_xor(s, o);
    s2 += __shfl_xor(s2, o);
  }
  float mu   = s * (1.0f / Cc);
  float var  = s2 * (1.0f / Cc) - mu * mu;
  float rstd = rsqrtf(var + 1e-5f);
  float4 g0 = *(const float4*)(g + c),  g1 = *(const float4*)(g + c + 4);
  float4 b0 = *(const float4*)(be + c), b1 = *(const float4*)(be + c + 4);
  float4 o0, o1;
  o0.x = (r0.x - mu) * rstd * g0.x + b0.x;
  o0.y = (r0.y - mu) * rstd * g0.y + b0.y;
  o0.z = (r0.z - mu) * rstd * g0.z + b0.z;
  o0.w = (r0.w - mu) * rstd * g0.w + b0.w;
  o1.x = (r1.x - mu) * rstd * g1.x + b1.x;
  o1.y = (r1.y - mu) * rstd * g1.y + b1.y;
  o1.z = (r1.z - mu) * rstd * g1.z + b1.z;
  o1.w = (r1.w - mu) * rstd * g1.w + b1.w;
  float* yp = y + (long)tok * Cc + c;
  *(float4*)(yp)     = o0;
  *(float4*)(yp + 4) = o1;
}

// ---------------------------------------------------------------------------
// Fused GEMM: C = A(MxK) @ B(KxN) + bias. Epilogue (compile-time MODE):
//   0: bias only             1: bias + exact GELU
//   2: bias + residual, scatter windowed row -> token order (proj epilogue)
//   3: bias + residual (same layout)         (mlp2 epilogue)
// 256 threads = 8 waves; block computes 16 rows x 128 cols (wave: 16x16 tile).
// A chunk (16 x 128) staged in LDS, stride 132 to avoid bank conflicts.
// ---------------------------------------------------------------------------
#define KCH 128
template <int MODE>
__global__ __launch_bounds__(256) void swin_gemm_kernel(
    const float* __restrict__ A, const float* __restrict__ Bm,
    const float* __restrict__ bias, float* __restrict__ Cout,
    const float* __restrict__ res, int M, int N, int K) {
  __shared__ float As[16][KCH + 4];
  int tid  = threadIdx.x;
  int wave = tid >> 5, lane = tid & 31;
  int m0   = blockIdx.y * 16;
  int n0   = blockIdx.x * 128 + wave * 16;
  int l15  = lane & 15;
  int kb   = (lane >> 4) << 1;   // 0 or 2
  int hi   = (lane >> 4) << 3;   // 0 or 8
  v8f acc = {};
  for (int k0 = 0; k0 < K; k0 += KCH) {
    {  // cooperative A-tile load: 2048 floats, 8 per thread (b128 loads)
      int r = tid >> 4;
      int c = (tid & 15) << 3;
      const float* ap = A + (long)(m0 + r) * K + k0 + c;
      float4 t0 = *(const float4*)(ap);
      float4 t1 = *(const float4*)(ap + 4);
      *(float4*)&As[r][c]     = t0;
      *(float4*)&As[r][c + 4] = t1;
    }
    __syncthreads();
    const float* bp = Bm + (long)k0 * N + n0 + l15;
    #pragma unroll 8
    for (int kk = 0; kk < KCH; kk += 4) {
      v2f a, b;
      a.x = As[l15][kk + kb];
      a.y = As[l15][kk + kb + 1];
      b.x = bp[(long)(kk + kb) * N];
      b.y = bp[(long)(kk + kb + 1) * N];
      acc = wmma_f32(a, b, acc);
    }
    __syncthreads();
  }
  int coln = n0 + l15;
  float bn = bias[coln];
  if (MODE == 0) {
    float* cp = Cout + (long)(m0 + hi) * N + coln;
    #pragma unroll
    for (int v = 0; v < 8; ++v) cp[(long)v * N] = acc[v] + bn;
  } else if (MODE == 1) {
    float* cp = Cout + (long)(m0 + hi) * N + coln;
    #pragma unroll
    for (int v = 0; v < 8; ++v) {
      float val = acc[v] + bn;
      cp[(long)v * N] = 0.5f * val * (1.0f + erff(val * 0.70710678118654752f));
    }
  } else if (MODE == 2) {
    // scatter: row m is a windowed token -> original token order (+ residual)
    #pragma unroll
    for (int v = 0; v < 8; ++v) {
      int m = m0 + v + hi;
      int win = m / NTOK, t = m % NTOK;
      int b = win / NWIN, w2 = win % NWIN;
      int wh = w2 >> 3, ww = w2 & 7;
      int i = t / WSZ, j = t % WSZ;
      int ho = (wh * WSZ + i + SHIFT) % Hh;
      int wo = (ww * WSZ + j + SHIFT) % Wy;
      long dst = ((long)b * (Hh * Wy) + ho * Wy + wo) * Cc + coln;
      Cout[dst] = res[dst] + acc[v] + bn;
    }
  } else {
    const float* rp = res + (long)(m0 + hi) * N + coln;
    float* cp = Cout + (long)(m0 + hi) * N + coln;
    #pragma unroll
    for (int v = 0; v < 8; ++v) cp[(long)v * N] = rp[(long)v * N] + acc[v] + bn;
  }
}

// ---------------------------------------------------------------------------
// Windowed attention: one block per (window, head). 128 threads = 4 waves.
// q/k/v (49x32, padded to 64x32) and S (64x64) live in LDS. Both matmuls
// (S = q k^T, O = P v) use V_WMMA_F32_16X16X4_F32. Relative-position bias and
// shift mask are computed analytically in the S epilogue.
// ---------------------------------------------------------------------------
__global__ __launch_bounds__(128) void swin_attn_kernel(
    const float* __restrict__ qkv, const float* __restrict__ rel_bias,
    float* __restrict__ out) {
  __shared__ float qs[64][HD + 1];
  __shared__ float ks[64][HD + 1];
  __shared__ float vs[64][HD + 1];
  __shared__ float Sm[64][68];
  int head = blockIdx.x & 7;
  int win  = blockIdx.x >> 3;
  int w2 = win % NWIN;
  int wh = w2 >> 3, ww = w2 & 7;
  int tid  = threadIdx.x;
  int wave = tid >> 5, lane = tid & 31;
  int l15 = lane & 15;
  int kb  = (lane >> 4) << 1;
  int hi  = (lane >> 4) << 3;

  const float* base = qkv + (long)win * NTOK * (3 * Cc) + head * HD;
  #pragma unroll
  for (int e0 = 0; e0 < 64 * HD; e0 += 128) {
    int e = e0 + tid;
    int t = e >> 5, d = e & 31;
    float qv = 0.f, kv = 0.f, vv = 0.f;
    if (t < NTOK) {
      const float* p = base + (long)t * (3 * Cc) + d;
      qv = p[0] * SCALE;
      kv = p[Cc];
      vv = p[2 * Cc];
    }
    qs[t][d] = qv; ks[t][d] = kv; vs[t][d] = vv;
  }
  __syncthreads();

  // S = q @ k^T  (+ rel bias + shift mask), wave handles rows [16w,16w+16)
  int m0 = wave * 16;
  #pragma unroll
  for (int nt = 0; nt < 4; ++nt) {
    int n0 = nt * 16;
    v8f acc = {};
    #pragma unroll
    for (int kk = 0; kk < HD; kk += 4) {
      v2f a, b;
      a.x = qs[m0 + l15][kk + kb];
      a.y = qs[m0 + l15][kk + kb + 1];
      b.x = ks[n0 + l15][kk + kb];
      b.y = ks[n0 + l15][kk + kb + 1];
      acc = wmma_f32(a, b, acc);
    }
    int n = n0 + l15;
    int in_ = n / WSZ, jn = n % WSZ;
    int idn = region_id(wh * WSZ + in_, ww * WSZ + jn);
    #pragma unroll
    for (int v = 0; v < 8; ++v) {
      int m = m0 + v + hi;
      float val = -1e30f;
      if (m < NTOK && n < NTOK) {
        int im = m / WSZ, jm = m % WSZ;
        int ridx = (im - in_ + WSZ - 1) * (2 * WSZ - 1) + (jm - jn + WSZ - 1);
        float bias = rel_bias[ridx * NHEADS + head];
        int idm = region_id(wh * WSZ + im, ww * WSZ + jm);
        val = acc[v] + bias + ((idm == idn) ? 0.0f : -100.0f);
      }
      Sm[m][n] = val;
    }
  }
  __syncthreads();

  // row softmax (threads 0..63, one row each)
  if (tid < 64) {
    float mx = -1e30f;
    #pragma unroll 8
    for (int n = 0; n < 64; ++n) mx = fmaxf(mx, Sm[tid][n]);
    float sum = 0.f;
    #pragma unroll 8
    for (int n = 0; n < 64; ++n) {
      float e = __expf(Sm[tid][n] - mx);
      Sm[tid][n] = e;
      sum += e;
    }
    float inv = 1.0f / sum;
    #pragma unroll 8
    for (int n = 0; n < 64; ++n) Sm[tid][n] *= inv;
  }
  __syncthreads();

  // O = P @ v : 64x32, wave handles rows [16w,16w+16), 2 n-tiles
  #pragma unroll
  for (int nt = 0; nt < 2; ++nt) {
    int n0 = nt * 16;
    v8f acc = {};
    #pragma unroll
    for (int kk = 0; kk < 64; kk += 4) {
      v2f a, b;
      a.x = Sm[m0 + l15][kk + kb];
      a.y = Sm[m0 + l15][kk + kb + 1];
      b.x = vs[kk + kb][n0 + l15];
      b.y = vs[kk + kb + 1][n0 + l15];
      acc = wmma_f32(a, b, acc);
    }
    float* op = out + ((long)win * NTOK + m0 + hi) * Cc + head * HD + n0 + l15;
    #pragma unroll
    for (int v = 0; v < 8; ++v) {
      int m = m0 + v + hi;
      if (m < NTOK) op[(long)v * Cc] = acc[v];
    }
  }
}

// ---------------------------------------------------------------------------
extern "C" void kernel_launch(void* const* d_in, const int* in_sizes, int n_in,
                              void* d_out, int out_size, void* d_ws,
                              size_t ws_size, hipStream_t stream) {
  const float* x       = (const float*)d_in[0];
  const float* qkv_w   = (const float*)d_in[1];
  const float* qkv_b   = (const float*)d_in[2];
  const float* proj_w  = (const float*)d_in[3];
  const float* proj_b  = (const float*)d_in[4];
  const float* rel_b   = (const float*)d_in[5];
  const float* ln1_g   = (const float*)d_in[6];
  const float* ln1_b   = (const float*)d_in[7];
  const float* ln2_g   = (const float*)d_in[8];
  const float* ln2_b   = (const float*)d_in[9];
  const float* mlp_w1  = (const float*)d_in[10];
  const float* mlp_b1  = (const float*)d_in[11];
  const float* mlp_w2  = (const float*)d_in[12];
  const float* mlp_b2  = (const float*)d_in[13];
  float* out = (float*)d_out;

  // workspace: buf0 (100352x256) | buf1 (100352x1024) | buf2 (100352x256)
  float* buf0 = (float*)d_ws;
  float* buf1 = buf0 + (size_t)TOKENS * Cc;
  float* buf2 = buf1 + (size_t)TOKENS * MLPD;

  // 1) LN1 + shift + window partition
  swin_ln_kernel<<<TOKENS / 8, 256, 0, stream>>>(x, ln1_g, ln1_b, buf0, 1);
  // 2) QKV GEMM (100352 x 768 x 256)
  swin_gemm_kernel<0><<<dim3(768 / 128, TOKENS / 16), 256, 0, stream>>>(
      buf0, qkv_w, qkv_b, buf1, nullptr, TOKENS, 3 * Cc, Cc);
  // 3) windowed attention (2048 windows x 8 heads)
  swin_attn_kernel<<<Bz * NWIN * NHEADS, 128, 0, stream>>>(buf1, rel_b, buf0);
  // 4) proj GEMM + residual + window-reverse/unshift scatter -> x1
  swin_gemm_kernel<2><<<dim3(256 / 128, TOKENS / 16), 256, 0, stream>>>(
      buf0, proj_w, proj_b, buf2, x, TOKENS, Cc, Cc);
  // 5) LN2
  swin_ln_kernel<<<TOKENS / 8, 256, 0, stream>>>(buf2, ln2_g, ln2_b, buf0, 0);
  // 6) MLP fc1 + exact GELU (100352 x 1024 x 256)
  swin_gemm_kernel<1><<<dim3(1024 / 128, TOKENS / 16), 256, 0, stream>>>(
      buf0, mlp_w1, mlp_b1, buf1, nullptr, TOKENS, MLPD, Cc);
  // 7) MLP fc2 + residual (100352 x 256 x 1024)
  swin_gemm_kernel<3><<<dim3(256 / 128, TOKENS / 16), 256, 0, stream>>>(
      buf1, mlp_w2, mlp_b2, out, buf2, TOKENS, Cc, MLPD);
}